// Model_14723147891262
// MI455X (gfx1250) — compile-verified
//
#include <hip/hip_runtime.h>
#include <hip/hip_bf16.h>
#include <math.h>

typedef float v2f __attribute__((ext_vector_type(2)));
typedef float v8f __attribute__((ext_vector_type(8)));

#define B_ 64
#define T_ 512
#define S_ 400
#define H_ 768
#define L_ 9
#define STRIDE_ 772   // 768 + 4: 16B-aligned rows, conflict-free b64 fragment loads

// ---------------- kernel 1: token-position and word-start maps (1 block / batch row)
__global__ __launch_bounds__(256) void map_kernel(
    const int* __restrict__ attn, const int* __restrict__ lens,
    int* __restrict__ tok_of_pos, int* __restrict__ nvalid, int* __restrict__ word_start)
{
  __shared__ int sA[T_];
  __shared__ int sB[T_];
  const int b = blockIdx.x;
  const int tid = threadIdx.x;

  // inclusive Hillis-Steele scan of attention mask over T
  for (int i = tid; i < T_; i += 256) sA[i] = attn[b*T_ + i];
  __syncthreads();
  int* src = sA; int* dst = sB;
  for (int off = 1; off < T_; off <<= 1) {
    for (int i = tid; i < T_; i += 256)
      dst[i] = src[i] + (i >= off ? src[i - off] : 0);
    __syncthreads();
    int* tmp = src; src = dst; dst = tmp;
  }
  for (int i = tid; i < T_; i += 256)
    if (attn[b*T_ + i] != 0) tok_of_pos[b*T_ + (src[i] - 1)] = i;
  if (tid == 0) nvalid[b] = src[T_ - 1];
  __syncthreads();

  // inclusive scan of ids_lens (padded to T_) -> exclusive word_start
  for (int i = tid; i < T_; i += 256) sA[i] = (i < S_) ? lens[b*S_ + i] : 0;
  __syncthreads();
  src = sA; dst = sB;
  for (int off = 1; off < T_; off <<= 1) {
    for (int i = tid; i < T_; i += 256)
      dst[i] = src[i] + (i >= off ? src[i - off] : 0);
    __syncthreads();
    int* tmp = src; src = dst; dst = tmp;
  }
  for (int i = tid; i < S_; i += 256)
    word_start[b*S_ + i] = (i == 0) ? 0 : src[i - 1];
}

// ---------------- kernel 2: fused mean-pool + WMMA classifier + softmax/NLL/argmax
// grid = (S/16, B), block = 256 (8 wave32s). One block handles 16 words of one row.
__global__ __launch_bounds__(256) void fused_kernel(
    const float* __restrict__ enc, const float* __restrict__ W, const float* __restrict__ bias,
    const int* __restrict__ lens, const int* __restrict__ labels,
    const int* __restrict__ tok_of_pos, const int* __restrict__ nvalid,
    const int* __restrict__ word_start,
    float* __restrict__ out, float* __restrict__ nll_out, float* __restrict__ mkf_out)
{
  extern __shared__ float sm[];
  float* smA = sm;                      // 16 x STRIDE_ : word means  (A, M=word)
  float* smB = sm + 16*STRIDE_;         // 16 x STRIDE_ : W^T padded  (B, row=N)
  float* smP = sm + 32*STRIDE_;         // 8 x 256      : per-wave partial C tiles
  float* smL = smP + 8*256;             // 256          : reduced logits

  const int b    = blockIdx.y;
  const int tile = blockIdx.x;
  const int tid  = threadIdx.x;
  const int wv   = tid >> 5;
  const int lane = tid & 31;
  const int hi   = (lane >> 4);         // 0: lanes 0-15, 1: lanes 16-31
  const int l16  = lane & 15;

  // stage W^T (padded N=16) into LDS, same layout as means
  for (int i = tid; i < 16*H_; i += 256) {
    int n = i / H_, k = i - n*H_;
    smB[n*STRIDE_ + k] = (n < L_) ? W[k*L_ + n] : 0.0f;
  }

  // --- pooling: wave wv owns words 2wv and 2wv+1; lane q-strided float4 loads (coalesced)
  const int nv = nvalid[b];
  for (int u = 0; u < 2; ++u) {
    const int wl = wv*2 + u;
    const int s  = tile*16 + wl;
    const int ln = lens[b*S_ + s];
    const int st = word_start[b*S_ + s];
    float4 acc[6];
#pragma unroll
    for (int q = 0; q < 6; ++q) acc[q] = make_float4(0.f, 0.f, 0.f, 0.f);
    for (int j = 0; j < ln; ++j) {
      const int p = st + j;
      if (p >= nv) break;                       // uniform within wave
      const int t = tok_of_pos[b*T_ + p];
      const float4* srcv = (const float4*)(enc + (size_t)(b*T_ + t) * H_);
#pragma unroll
      for (int q = 0; q < 6; ++q) {
        float4 v = srcv[lane + 32*q];
        acc[q].x += v.x; acc[q].y += v.y; acc[q].z += v.z; acc[q].w += v.w;
      }
    }
    const float sc = 1.0f / (float)((ln > 0) ? ln : 1);
    float4* dstv = (float4*)(smA + wl*STRIDE_);
#pragma unroll
    for (int q = 0; q < 6; ++q) {
      float4 v; v.x = acc[q].x*sc; v.y = acc[q].y*sc; v.z = acc[q].z*sc; v.w = acc[q].w*sc;
      dstv[lane + 32*q] = v;
    }
  }
  __syncthreads();

  // --- WMMA f32 16x16x4: each wave takes a K-slice of 96 (24 steps)
  v8f cacc = {0.f, 0.f, 0.f, 0.f, 0.f, 0.f, 0.f, 0.f};
  const float* arow = smA + l16*STRIDE_ + hi*2;   // A: lane=M, VGPR0/1 = K{0,1}|{2,3}
  const float* brow = smB + l16*STRIDE_ + hi*2;   // B: lane=N, same K striping
  const int kbase = wv * 96;
#pragma unroll 4
  for (int i = 0; i < 24; ++i) {
    const int k = kbase + i*4;
    v2f a  = *(const v2f*)(arow + k);
    v2f bf = *(const v2f*)(brow + k);
    cacc = __builtin_amdgcn_wmma_f32_16x16x4_f32(
        false, a, false, bf, (short)0, cacc, false, false);
  }
  // deterministic cross-wave reduce: spill per-wave tiles, fixed-order sum
#pragma unroll
  for (int r = 0; r < 8; ++r) {
    const int m = r + hi*8;                       // C layout: VGPR r -> M = r (+8 hi)
    smP[wv*256 + m*16 + l16] = cacc[r];
  }
  __syncthreads();
  {
    const int m = tid >> 4, n = tid & 15;
    float v = 0.0f;
#pragma unroll
    for (int w8 = 0; w8 < 8; ++w8) v += smP[w8*256 + m*16 + n];
    v += (n < L_) ? bias[n] : 0.0f;
    smL[m*16 + n] = v;
  }
  __syncthreads();

  // --- per-word softmax / NLL / argmax (first 16 threads)
  if (tid < 16) {
    const int m = tid;
    const int s = tile*16 + m;
    float mx = smL[m*16];
    int am = 0;
    for (int n = 1; n < L_; ++n) {
      float v = smL[m*16 + n];
      if (v > mx) { mx = v; am = n; }             // first-max tie-break like argmax
    }
    float se = 0.0f;
    for (int n = 0; n < L_; ++n) se += expf(smL[m*16 + n] - mx);
    const int lab = labels[b*S_ + s];
    const float nll = -(smL[m*16 + lab] - mx - logf(se));
    const float mk = (lens[b*S_ + s] > 0) ? 1.0f : 0.0f;
    nll_out[b*S_ + s] = nll * mk;
    mkf_out[b*S_ + s] = mk;
    out[1 + b*S_ + s] = (float)am;
  }
}

// ---------------- kernel 3: deterministic loss reduction
__global__ __launch_bounds__(256) void reduce_kernel(
    const float* __restrict__ nll, const float* __restrict__ mkf, float* __restrict__ out)
{
  __shared__ float s1[256];
  __shared__ float s2[256];
  const int tid = threadIdx.x;
  float a = 0.0f, c = 0.0f;
  for (int i = tid; i < B_*S_; i += 256) { a += nll[i]; c += mkf[i]; }
  s1[tid] = a; s2[tid] = c;
  __syncthreads();
  for (int off = 128; off > 0; off >>= 1) {
    if (tid < off) { s1[tid] += s1[tid + off]; s2[tid] += s2[tid + off]; }
    __syncthreads();
  }
  if (tid == 0) out[0] = s1[0] / s2[0];
}

extern "C" void kernel_launch(void* const* d_in, const int* in_sizes, int n_in,
                              void* d_out, int out_size, void* d_ws, size_t ws_size,
                              hipStream_t stream) {
  (void)in_sizes; (void)n_in; (void)out_size; (void)ws_size;
  const float* enc  = (const float*)d_in[0];
  const float* W    = (const float*)d_in[1];
  const float* bias = (const float*)d_in[2];
  const int*   attn = (const int*)d_in[3];
  const int*   lens = (const int*)d_in[4];
  const int*   labs = (const int*)d_in[5];
  float* out = (float*)d_out;

  char* ws = (char*)d_ws;
  int*   tok_of_pos = (int*)ws;    ws += (size_t)B_*T_*sizeof(int);
  int*   nvalid     = (int*)ws;    ws += 256;
  int*   word_start = (int*)ws;    ws += (size_t)B_*S_*sizeof(int);
  float* nll        = (float*)ws;  ws += (size_t)B_*S_*sizeof(float);
  float* mkf        = (float*)ws;

  map_kernel<<<B_, 256, 0, stream>>>(attn, lens, tok_of_pos, nvalid, word_start);

  dim3 grid(S_/16, B_);
  size_t shmem = (size_t)(32*STRIDE_ + 8*256 + 256) * sizeof(float); // ~105.5 KB
  fused_kernel<<<grid, 256, shmem, stream>>>(enc, W, bias, lens, labs,
      tok_of_pos, nvalid, word_start, out, nll, mkf);

  reduce_kernel<<<1, 256, 0, stream>>>(nll, mkf, out);
}